// RecurrentNetworkAgg_32246614458490
// MI455X (gfx1250) — compile-verified
//
#include <hip/hip_runtime.h>
#include <cmath>

typedef __attribute__((ext_vector_type(2))) float v2f;
typedef __attribute__((ext_vector_type(8))) float v8f;

#if __has_builtin(__builtin_amdgcn_tanhf)
#define FAST_TANH(x) __builtin_amdgcn_tanhf(x)   // V_TANH_F32 (single trans op)
#else
#define FAST_TANH(x) tanhf(x)
#endif

#define NOUT 3
#define NN   67        // N_OUT + N_HID, fixed by the reference
#define NPAD 80        // neuron dim padded to 5 tiles of 16
#define KPAD 68        // recurrent K padded to 17 tiles of 4
#define WAVES 8
#define RPW  16        // rows per wave (one 16-row WMMA M-tile)
#define RPB  (WAVES * RPW)

__global__ __launch_bounds__(256) void rnn_wmma_kernel(
    const float* __restrict__ x,     // (M, 4)
    const float* __restrict__ Win,   // (4, NN)
    const float* __restrict__ Wrec,  // (NN, NN)
    const float* __restrict__ bias,  // (NN,)
    const float* __restrict__ resp,  // (NN,)
    const float* __restrict__ has,   // (NN,)
    const int*   __restrict__ n_passes,
    float* __restrict__ out,         // (M, 3)
    int M)
{
    constexpr int Nn = NN;           // compile-time: staging divides fold to mul-shift

    __shared__ float sWin[NPAD * 4];          // transposed [n][k]
    __shared__ float sWrec[NPAD * KPAD];      // transposed [n][k], zero-padded
    __shared__ float sVec[3 * NPAD];          // bias | response | has_in
    __shared__ float sAct[WAVES * RPW * KPAD];// per-wave activation tiles [m][k]

    const int tid = threadIdx.x;

    // ---- phase 1: zero padded weight arrays (LDS only) ----
    for (int idx = tid; idx < NPAD * KPAD; idx += 256) sWrec[idx] = 0.0f;
    for (int idx = tid; idx < NPAD * 4;    idx += 256) sWin[idx]  = 0.0f;
    for (int idx = tid; idx < 3 * NPAD;    idx += 256) sVec[idx]  = 0.0f;
    __syncthreads();

    // ---- phase 2: fill from global in MEMORY order (coalesced reads),
    //      scatter the transpose into LDS ----
    for (int idx = tid; idx < Nn * Nn; idx += 256) {
        int k = idx / Nn, n = idx - k * Nn;   // Wrec[k][n], row-major
        sWrec[n * KPAD + k] = Wrec[idx];
    }
    for (int idx = tid; idx < 4 * Nn; idx += 256) {
        int k = idx / Nn, n = idx - k * Nn;   // Win[k][n], row-major
        sWin[n * 4 + k] = Win[idx];
    }
    for (int idx = tid; idx < Nn; idx += 256) {
        sVec[idx]            = bias[idx];
        sVec[NPAD + idx]     = resp[idx];
        sVec[2 * NPAD + idx] = has[idx];      // pad stays 0 => pad activ 0
    }
    __syncthreads();

    const int lane  = tid & 31;
    const int wave  = tid >> 5;
    const int mrow  = lane & 15;     // row (A) / column (B,C) within tile
    const int khalf = lane >> 4;     // which K-pair half this lane holds
    const long long row_base = ((long long)blockIdx.x * WAVES + wave) * RPW;
    if (row_base >= M) return;       // wave-uniform exit (EXEC stays all-1s)
    const long long nrows = ((long long)M - row_base) < RPW
                              ? ((long long)M - row_base) : (long long)RPW;

    // ---- A fragment of x: 16x4 f32 tile, K = 4 exactly; coalesced b64 ----
    const int mr = (mrow < nrows) ? mrow : 0;
    v2f a_in = *(const v2f*)(x + (row_base + mr) * 4 + 2 * khalf);

    // ---- cin[n] = x @ W_in : one V_WMMA_F32_16X16X4_F32 per N-tile ----
    v8f cin[5];
    v8f zero = {};
    #pragma unroll
    for (int n = 0; n < 5; ++n) {
        v2f b = *(const v2f*)&sWin[(n * 16 + mrow) * 4 + 2 * khalf];
        cin[n] = __builtin_amdgcn_wmma_f32_16x16x4_f32(
            false, a_in, false, b, (short)0, zero, false, false);
    }

    const int np = *n_passes;
    float* aw = &sAct[wave * RPW * KPAD];                // private per-wave tile
    const float* arow = aw + mrow * KPAD + 2 * khalf;    // A-fragment base

    // per-lane elementwise constants (C layout: lane -> column ncol)
    float bi[5], re[5], ha[5];
    #pragma unroll
    for (int n = 0; n < 5; ++n) {
        int ncol = n * 16 + mrow;
        bi[n] = sVec[ncol];
        re[n] = sVec[NPAD + ncol];
        ha[n] = sVec[2 * NPAD + ncol];
    }

    // ---- pass 0: activs == 0 so agg == x@W_in; no recurrent matmul, and the
    //      elementwise store initializes every K column (pad col 67 -> 0) ----
    #pragma unroll
    for (int n = 0; n < 5; ++n) {
        const int ncol = n * 16 + mrow;
        #pragma unroll
        for (int v = 0; v < 8; ++v) {
            float val = ha[n] * FAST_TANH(fmaf(re[n], cin[n][v], bi[n]));
            int r = v + 8 * khalf;
            if (ncol < KPAD) aw[r * KPAD + ncol] = val;
        }
    }

    // ---- passes 1..np-1: agg = x@W_in + activs@W_rec ----
    for (int p = 1; p < np; ++p) {
        v8f c[5];
        #pragma unroll
        for (int n = 0; n < 5; ++n) c[n] = cin[n];
        #pragma unroll
        for (int kt = 0; kt < KPAD / 4; ++kt) {
            v2f a = *(const v2f*)(arow + 4 * kt);        // ds_load_b64
            #pragma unroll
            for (int n = 0; n < 5; ++n) {
                // volatile: keep B fragments as in-loop LDS loads instead of
                // pinning ~170 VGPRs of hoisted weights across the pass loop
                v2f b = *(volatile const v2f*)&sWrec[(n * 16 + mrow) * KPAD + 4 * kt + 2 * khalf];
                c[n] = __builtin_amdgcn_wmma_f32_16x16x4_f32(
                    false, a, false, b, (short)0, c[n], false, false);
            }
        }
        #pragma unroll
        for (int n = 0; n < 5; ++n) {
            const int ncol = n * 16 + mrow;
            #pragma unroll
            for (int v = 0; v < 8; ++v) {
                float val = ha[n] * FAST_TANH(fmaf(re[n], c[n][v], bi[n]));
                int r = v + 8 * khalf;
                if (ncol < KPAD) aw[r * KPAD + ncol] = val;
            }
        }
        // per-wave LDS ops are in-order; no block barrier needed
    }

    // ---- emit out[:, 0:3]: 48 contiguous floats per wave ----
    const long long ob = row_base * NOUT;
    const int lim = (int)nrows * NOUT;
    for (int i = lane; i < lim; i += 32) {
        int m = i / NOUT, nn = i - NOUT * m;
        out[ob + i] = aw[m * KPAD + nn];
    }
}

extern "C" void kernel_launch(void* const* d_in, const int* in_sizes, int n_in,
                              void* d_out, int out_size, void* d_ws, size_t ws_size,
                              hipStream_t stream) {
    const float* x    = (const float*)d_in[0];
    const float* Win  = (const float*)d_in[1];
    const float* Wrec = (const float*)d_in[2];
    const float* bias = (const float*)d_in[3];
    const float* resp = (const float*)d_in[4];
    const float* has  = (const float*)d_in[5];
    const int*   np   = (const int*)d_in[6];
    float* out = (float*)d_out;

    const int M = in_sizes[0] / 4;    // rows = inputs.size / N_IN
    const int blocks = (M + RPB - 1) / RPB;
    rnn_wmma_kernel<<<blocks, 256, 0, stream>>>(x, Win, Wrec, bias, resp, has,
                                                np, out, M);
}